// ProbAttention_6768868458798
// MI455X (gfx1250) — compile-verified
//
#include <hip/hip_runtime.h>
#include <math.h>

// ProbSparse attention (Informer) for B=4, L=2048, H=8, D=64, S=u=40 (FACTOR=5).
// Key observation: the reference's causal mask is indexed by top-k RANK, not by
// query position -> every selected query attends only to keys 0..39. The dense
// 40x2048 attention collapses to 40x48 (padded); the pipeline is dominated by
// the sampled-score gather (~670MB, L2-resident K) and the V cumsum (~50MB).
// GEMMs use V_WMMA_F32_16X16X4_F32.

typedef __attribute__((ext_vector_type(2))) float v2f;
typedef __attribute__((ext_vector_type(8))) float v8f;

#define B_ 4
#define L_ 2048
#define H_ 8
#define D_ 64
#define S_ 40
#define U_ 40
#define UPAD 48
#define NCHUNK 64
#define CHUNK 32   // L_/NCHUNK

// ---------------------------------------------------------------------------
// Kernel 1: M[b,h,q] = max_s(Q[q].K[idx[q,s]]) - sum_s(...)/L
// One wave per (b,h,q); lanes cover D as float2; shfl reduction per sample.
// ---------------------------------------------------------------------------
__global__ __launch_bounds__(256) void k_sample_scores(
    const float* __restrict__ Q, const float* __restrict__ K,
    const int* __restrict__ idx, float* __restrict__ Mout)
{
  int wid  = blockIdx.x * 8 + (threadIdx.x >> 5);   // (b*H+h)*L + q
  int lane = threadIdx.x & 31;
  int q  = wid % L_;
  int bh = wid / L_;
  int b = bh / H_, h = bh % H_;
  const float* qrow = Q + (((size_t)(b * L_ + q) * H_ + h) * D_);
  float2 qv = *(const float2*)(qrow + lane * 2);
  float mx = -INFINITY, sm = 0.f;
  for (int s = 0; s < S_; ++s) {
    int ki = idx[q * S_ + s];
    const float* krow = K + (((size_t)(b * L_ + ki) * H_ + h) * D_);
    float2 kv = *(const float2*)(krow + lane * 2);
    float d = qv.x * kv.x + qv.y * kv.y;
    #pragma unroll
    for (int off = 16; off; off >>= 1) d += __shfl_xor(d, off, 32);
    mx = fmaxf(mx, d);
    sm += d;
  }
  if (lane == 0) Mout[bh * L_ + q] = mx - sm * (1.0f / (float)L_);
}

// ---------------------------------------------------------------------------
// Kernel 2: per (b,h): top-40 of M (descending, ties -> lowest index, matching
// jax.lax.top_k). Emits ordered indices + inverse rank map (-1 = not selected).
// ---------------------------------------------------------------------------
__global__ __launch_bounds__(256) void k_topk(
    const float* __restrict__ Mval, int* __restrict__ top, int* __restrict__ rank)
{
  __shared__ float vals[L_];
  __shared__ float rv[256];
  __shared__ int   ri[256];
  int bh = blockIdx.x, tid = threadIdx.x;
  for (int i = tid; i < L_; i += 256) {
    vals[i] = Mval[bh * L_ + i];
    rank[bh * L_ + i] = -1;
  }
  __syncthreads();
  for (int r = 0; r < U_; ++r) {
    float bv = -INFINITY; int bi = 0x7fffffff;
    for (int i = tid; i < L_; i += 256) {   // ascending i: strict > keeps lowest idx on ties
      float v = vals[i];
      if (v > bv) { bv = v; bi = i; }
    }
    rv[tid] = bv; ri[tid] = bi;
    __syncthreads();
    for (int off = 128; off; off >>= 1) {
      if (tid < off) {
        float ov = rv[tid + off]; int oi = ri[tid + off];
        if (ov > rv[tid] || (ov == rv[tid] && oi < ri[tid])) { rv[tid] = ov; ri[tid] = oi; }
      }
      __syncthreads();
    }
    if (tid == 0) {
      int bidx = ri[0];
      top[bh * U_ + r] = bidx;
      rank[bh * L_ + bidx] = r;
      vals[bidx] = -INFINITY;
    }
    __syncthreads();
  }
}

// ---------------------------------------------------------------------------
// Kernel 3: per (b,h): attention for the 40 selected queries (padded to 48
// rows = 3 waves x 16). Because the mask is rank-indexed (row i sees keys
// j<=i, i<40), only key chunks c=0..2 (keys 0..47) can be unmasked; all later
// chunks contribute exactly zero, so the flash loop stops at c<3 (bit-exact).
//
// WMMA f32 layouts used (wave32):
//   A 16x4:  lane m (0-15)/(16-31) -> M=m&15; reg0 = K(+0/+2), reg1 = K(+1/+3)
//   B 4x16:  lane n -> N=n&15;               reg0 = K(+0/+2), reg1 = K(+1/+3)
//   C 16x16: reg r, lanes 0-15 -> M=r, lanes 16-31 -> M=r+8, N=lane&15
// Row stats (m,l) live per (r, lane-half), matching the C layout; reductions
// are shfl_xor over the 16-lane half. P bounces through LDS into A layout.
// ---------------------------------------------------------------------------
__global__ __launch_bounds__(96) void k_attn(
    const float* __restrict__ Q, const float* __restrict__ K,
    const float* __restrict__ V, const int* __restrict__ top,
    float* __restrict__ upd)
{
  __shared__ float Qs[UPAD * D_];
  __shared__ float Ks[16 * D_];
  __shared__ float Vs[16 * D_];
  __shared__ float Ps[3 * 16 * 16];

  int bh = blockIdx.x; int b = bh / H_, h = bh % H_;
  int tid  = threadIdx.x;
  int wave = tid >> 5, lane = tid & 31;
  int mrow = lane & 15;
  int koff = (lane < 16) ? 0 : 2;
  int half = lane >> 4;

  for (int i = tid; i < UPAD * D_; i += 96) {
    int r = i >> 6, d = i & 63;
    float v = 0.f;
    if (r < U_) {
      int qi = top[bh * U_ + r];
      v = Q[(((size_t)(b * L_ + qi)) * H_ + h) * D_ + d];
    }
    Qs[i] = v;
  }
  __syncthreads();

  v8f o0 = {}, o1 = {}, o2 = {}, o3 = {};
  float m_run[8], l_run[8];
  #pragma unroll
  for (int r = 0; r < 8; ++r) { m_run[r] = -INFINITY; l_run[r] = 0.f; }

  for (int c = 0; c < UPAD / 16; ++c) {     // keys 0..47 cover every unmasked j
    for (int i = tid; i < 16 * D_; i += 96) {
      int j = i >> 6, d = i & 63;
      size_t g = (((size_t)(b * L_ + c * 16 + j)) * H_ + h) * D_ + d;
      Ks[i] = K[g];
      Vs[i] = V[g];
    }
    __syncthreads();

    // S = Qw(16x64) @ Kc^T(64x16), accumulated 4 K-dims at a time.
    v8f s = {};
    #pragma unroll
    for (int k0 = 0; k0 < D_; k0 += 4) {
      v2f a, bb;
      a.x  = Qs[(wave * 16 + mrow) * D_ + k0 + koff];
      a.y  = Qs[(wave * 16 + mrow) * D_ + k0 + koff + 1];
      bb.x = Ks[mrow * D_ + k0 + koff];
      bb.y = Ks[mrow * D_ + k0 + koff + 1];
      s = __builtin_amdgcn_wmma_f32_16x16x4_f32(false, a, false, bb, (short)0, s,
                                                false, false);
    }

    // Online softmax update (scale=1/sqrt(64); mask: key j > rank row => -inf)
    int jglob = c * 16 + mrow;
    #pragma unroll
    for (int r = 0; r < 8; ++r) {
      int Mloc = r + half * 8;
      int grow = wave * 16 + Mloc;
      float sv = s[r] * 0.125f;
      if (jglob > grow) sv = -INFINITY;
      float mxv = sv;
      #pragma unroll
      for (int off = 8; off; off >>= 1) mxv = fmaxf(mxv, __shfl_xor(mxv, off, 16));
      float mnew = fmaxf(m_run[r], mxv);
      float sc = __expf(m_run[r] - mnew);
      float p  = __expf(sv - mnew);
      float rs = p;
      #pragma unroll
      for (int off = 8; off; off >>= 1) rs += __shfl_xor(rs, off, 16);
      l_run[r] = l_run[r] * sc + rs;
      m_run[r] = mnew;
      o0[r] *= sc; o1[r] *= sc; o2[r] *= sc; o3[r] *= sc;
      Ps[wave * 256 + Mloc * 16 + mrow] = p;   // per-wave region, row-major
    }
    // same-wave LDS store->load is in-order; no block barrier needed here

    // O += P(16x16) @ Vc(16x64): 4 K-steps x 4 output column tiles.
    #pragma unroll
    for (int k0 = 0; k0 < 16; k0 += 4) {
      v2f a;
      a.x = Ps[wave * 256 + mrow * 16 + k0 + koff];
      a.y = Ps[wave * 256 + mrow * 16 + k0 + koff + 1];
      v2f b0, b1, b2, b3;
      b0.x = Vs[(k0 + koff) * D_ +  0 + mrow]; b0.y = Vs[(k0 + koff + 1) * D_ +  0 + mrow];
      b1.x = Vs[(k0 + koff) * D_ + 16 + mrow]; b1.y = Vs[(k0 + koff + 1) * D_ + 16 + mrow];
      b2.x = Vs[(k0 + koff) * D_ + 32 + mrow]; b2.y = Vs[(k0 + koff + 1) * D_ + 32 + mrow];
      b3.x = Vs[(k0 + koff) * D_ + 48 + mrow]; b3.y = Vs[(k0 + koff + 1) * D_ + 48 + mrow];
      o0 = __builtin_amdgcn_wmma_f32_16x16x4_f32(false, a, false, b0, (short)0, o0, false, false);
      o1 = __builtin_amdgcn_wmma_f32_16x16x4_f32(false, a, false, b1, (short)0, o1, false, false);
      o2 = __builtin_amdgcn_wmma_f32_16x16x4_f32(false, a, false, b2, (short)0, o2, false, false);
      o3 = __builtin_amdgcn_wmma_f32_16x16x4_f32(false, a, false, b3, (short)0, o3, false, false);
    }
    __syncthreads();  // protect Ks/Vs before next chunk overwrite
  }

  #pragma unroll
  for (int r = 0; r < 8; ++r) {
    int grow = wave * 16 + r + half * 8;
    if (grow < U_) {
      float inv = 1.0f / l_run[r];
      size_t base = ((size_t)bh * U_ + grow) * D_;
      upd[base +  0 + mrow] = o0[r] * inv;
      upd[base + 16 + mrow] = o1[r] * inv;
      upd[base + 32 + mrow] = o2[r] * inv;
      upd[base + 48 + mrow] = o3[r] * inv;
    }
  }
}

// ---------------------------------------------------------------------------
// Kernels 4-6: chunked cumsum of V along L (per b,h,d) + scatter of upd rows.
// NCHUNK=64 -> 2048 blocks per streaming pass (4096 waves) to feed 23.3 TB/s.
// ---------------------------------------------------------------------------
__global__ __launch_bounds__(64) void k_partial(const float* __restrict__ V,
                                                float* __restrict__ part)
{
  int blk = blockIdx.x;                 // bh*NCHUNK + chunk
  int bh = blk / NCHUNK, ck = blk % NCHUNK;
  int b = bh / H_, h = bh % H_;
  int d = threadIdx.x;
  float acc = 0.f;
  int l0 = ck * CHUNK;
  for (int l = l0; l < l0 + CHUNK; ++l)
    acc += V[(((size_t)(b * L_ + l)) * H_ + h) * D_ + d];
  part[(size_t)blk * D_ + d] = acc;
}

__global__ __launch_bounds__(64) void k_scan(float* __restrict__ part)
{
  int bh = blockIdx.x, d = threadIdx.x;
  float run = 0.f;
  for (int ck = 0; ck < NCHUNK; ++ck) {
    size_t off = ((size_t)bh * NCHUNK + ck) * D_ + d;
    float t = part[off];
    part[off] = run;                    // exclusive prefix
    run += t;
  }
}

__global__ __launch_bounds__(64) void k_out(
    const float* __restrict__ V, const float* __restrict__ part,
    const int* __restrict__ rank, const float* __restrict__ upd,
    float* __restrict__ out)
{
  int blk = blockIdx.x;
  int bh = blk / NCHUNK, ck = blk % NCHUNK;
  int b = bh / H_, h = bh % H_;
  int d = threadIdx.x;
  float acc = part[(size_t)blk * D_ + d];
  int l0 = ck * CHUNK;
  for (int l = l0; l < l0 + CHUNK; ++l) {
    size_t g = (((size_t)(b * L_ + l)) * H_ + h) * D_ + d;
    acc += V[g];                                   // inclusive cumsum
    int r = rank[bh * L_ + l];
    out[g] = (r >= 0) ? upd[((size_t)bh * U_ + r) * D_ + d] : acc;
  }
}

// ---------------------------------------------------------------------------
extern "C" void kernel_launch(void* const* d_in, const int* in_sizes, int n_in,
                              void* d_out, int out_size, void* d_ws, size_t ws_size,
                              hipStream_t stream)
{
  const float* Q  = (const float*)d_in[0];
  const float* K  = (const float*)d_in[1];
  const float* V  = (const float*)d_in[2];
  const int* idx  = (const int*)d_in[3];
  float* out = (float*)d_out;

  char* ws = (char*)d_ws;
  float* ws_M    = (float*)(ws);              //  65536 f  (B*H*L)
  int*   ws_top  = (int*)  (ws + 262144);     //   1280 i  (B*H*U)
  int*   ws_rank = (int*)  (ws + 267264);     //  65536 i  (B*H*L)
  float* ws_upd  = (float*)(ws + 529408);     //  81920 f  (B*H*U*D)
  float* ws_part = (float*)(ws + 857088);     // 131072 f  (B*H*NCHUNK*D)
  (void)in_sizes; (void)n_in; (void)out_size; (void)ws_size;

  k_sample_scores<<<B_ * H_ * L_ / 8, 256, 0, stream>>>(Q, K, idx, ws_M);
  k_topk<<<B_ * H_, 256, 0, stream>>>(ws_M, ws_top, ws_rank);
  k_attn<<<B_ * H_, 96, 0, stream>>>(Q, K, V, ws_top, ws_upd);
  k_partial<<<B_ * H_ * NCHUNK, 64, 0, stream>>>(V, ws_part);
  k_scan<<<B_ * H_, 64, 0, stream>>>(ws_part);
  k_out<<<B_ * H_ * NCHUNK, 64, 0, stream>>>(V, ws_part, ws_rank, ws_upd, out);
}